// CnaHGNN_41068477285176
// MI455X (gfx1250) — compile-verified
//
#include <hip/hip_runtime.h>
#include <hip/hip_bf16.h>

// ---------------------------------------------------------------------------
// CnaHGNN forward for MI455X (gfx1250, wave32).
// Dense GEMMs use V_WMMA_F32_16X16X4_F32 (f32 in / f32 acc = reference
// numerics).  All guarded loads replaced by index-clamping so GEMM inner loops
// are branch-free (no exec-mask churn, b64 loads, dense WMMA runs).
// spmm = edge-parallel gather + global_atomic_add_f32 scatter (51MB feature
// matrices are L2-resident on the 192MB L2).  l2norm fused into the output
// gathers; sigmoid/MSE loss reduced in-register and never materialized.
// ---------------------------------------------------------------------------

#define NFEAT   128
#define NATTRI  500
#define NNODES  100000
#define BETA_POS 2.0f
#define BETA_NEG 0.1f
#define EPSN     1e-12f

typedef __attribute__((ext_vector_type(2))) float v2f;
typedef __attribute__((ext_vector_type(8))) float v8f;

__device__ __forceinline__ v8f wmma_f32x4(v2f a, v2f b, v8f c) {
  // D = A(16x4,f32) * B(4x16,f32) + C(16x16,f32)   -> v_wmma_f32_16x16x4_f32
  return __builtin_amdgcn_wmma_f32_16x16x4_f32(
      /*neg_a=*/false, a, /*neg_b=*/false, b,
      /*c_mod=*/(short)0, c, /*reuse_a=*/false, /*reuse_b=*/false);
}

__device__ __forceinline__ void atomic_add_f32(float* p, float v) {
  __hip_atomic_fetch_add(p, v, __ATOMIC_RELAXED, __HIP_MEMORY_SCOPE_AGENT);
}

__device__ __forceinline__ int global_wave() {
  return blockIdx.x * (blockDim.x >> 5) + (threadIdx.x >> 5);
}

// --------------------------------------------------------------------------
// loss = BETA_POS*mean(mse*(H==1)) + BETA_NEG*mean(mse*(H==0))
// One wave per 16-node tile.  The emb_node fragments for K=128 are held in
// 64 VGPRs and reused across all 32 attr-tiles -> emb_node is read exactly
// once, and each attr-tile is a dense run of 32 branch-free WMMAs.
// Tail rows (500..511) are handled by clamping the load index; their garbage
// C rows are discarded in the reduction (never fed to WMMA masking).
// --------------------------------------------------------------------------
__global__ void recon_loss_kernel(const float* __restrict__ emb_attr,
                                  const float* __restrict__ emb_node,
                                  const float* __restrict__ H,
                                  float* __restrict__ acc /* [pos,neg] */) {
  const int nNodeTiles = NNODES / 16;            // 6250
  const int tileN = global_wave();
  if (tileN >= nNodeTiles) return;
  const int lane = threadIdx.x & 31;
  const int mrow = lane & 15;
  const int kh   = lane >> 4;

  // Cache B = emb_node^T fragments for the whole K=128 in registers.
  const float* Bp = emb_node + (size_t)(tileN * 16 + mrow) * NFEAT + 2 * kh;
  v2f breg[32];
#pragma unroll
  for (int i = 0; i < 32; ++i)
    breg[i] = *reinterpret_cast<const v2f*>(Bp + 4 * i);

  const int nCol = tileN * 16 + mrow;            // C: N = lane&15
  float posS = 0.0f, negS = 0.0f;

  for (int ta = 0; ta < (NATTRI + 15) / 16; ++ta) {
    const int aRow = ta * 16 + mrow;
    const int aClamp = aRow < NATTRI ? aRow : (NATTRI - 1);   // branch-free tail
    const float* Ap = emb_attr + (size_t)aClamp * NFEAT + 2 * kh;

    v8f c = {};
#pragma unroll
    for (int i = 0; i < 32; ++i) {
      const v2f a = *reinterpret_cast<const v2f*>(Ap + 4 * i);
      c = wmma_f32x4(a, breg[i], c);
    }

#pragma unroll
    for (int v = 0; v < 8; ++v) {
      const int m = ta * 16 + v + 8 * kh;        // C: M = v + 8*(lane>>4)
      const int mc = m < NATTRI ? m : (NATTRI - 1);
      const float h = H[(size_t)mc * NNODES + nCol];
      const float s = 1.0f / (1.0f + __expf(-c[v]));
      const float d = h - s;
      const float e2 = d * d;
      const bool valid = (m < NATTRI);
      posS += (valid && h == 1.0f) ? e2 : 0.0f;
      negS += (valid && h != 1.0f) ? e2 : 0.0f;
    }
  }

  for (int off = 16; off > 0; off >>= 1) {
    posS += __shfl_down(posS, off, 32);
    negS += __shfl_down(negS, off, 32);
  }
  if (lane == 0) {
    atomic_add_f32(&acc[0], posS);
    atomic_add_f32(&acc[1], negS);
  }
}

// --------------------------------------------------------------------------
// agg = H^T @ emb_attr : (100000 x 500) @ (500 x 128).  One wave per 16-node
// block computes all 8 feature tiles (8 x v8f accumulators), so each H
// A-fragment (the expensive, 200MB side) feeds 8 WMMAs.
// --------------------------------------------------------------------------
__global__ void agg_kernel(const float* __restrict__ H,
                           const float* __restrict__ emb_attr,
                           float* __restrict__ agg) {
  const int tileN = global_wave();
  if (tileN >= NNODES / 16) return;
  const int lane = threadIdx.x & 31;
  const int mrow = lane & 15;
  const int kh   = lane >> 4;
  const int node = tileN * 16 + mrow;

  v8f c[8];
#pragma unroll
  for (int t = 0; t < 8; ++t) c[t] = (v8f){};

  for (int k0 = 0; k0 < NATTRI; k0 += 4) {       // 500 % 4 == 0
    const int ka = k0 + 2 * kh;
    v2f a;
    a.x = H[(size_t)ka       * NNODES + node];   // A[m][k] = H[k][m]
    a.y = H[(size_t)(ka + 1) * NNODES + node];
    const float* W0r = emb_attr + (size_t)ka * NFEAT + mrow;
#pragma unroll
    for (int t = 0; t < 8; ++t) {
      v2f b;
      b.x = W0r[t * 16];
      b.y = W0r[NFEAT + t * 16];
      c[t] = wmma_f32x4(a, b, c[t]);
    }
  }

#pragma unroll
  for (int t = 0; t < 8; ++t) {
#pragma unroll
    for (int v = 0; v < 8; ++v) {
      const int m = tileN * 16 + v + 8 * kh;
      agg[(size_t)m * NFEAT + t * 16 + mrow] = c[t][v];
    }
  }
}

// --------------------------------------------------------------------------
// out = [A0 | A1] @ W + bias  (A1 may be null when K == 128; the concat is
// fused by splitting the K-loop into two wave-uniform halves -> no per-lane
// pointer selects).  One wave per 16-node block, all 8 feature tiles.
// --------------------------------------------------------------------------
__global__ void linear_kernel(const float* __restrict__ A0,
                              const float* __restrict__ A1,
                              const float* __restrict__ W,     // K x NFEAT
                              const float* __restrict__ bias,  // NFEAT
                              float* __restrict__ out,         // NNODES x NFEAT
                              int K) {
  const int tileN = global_wave();
  if (tileN >= NNODES / 16) return;
  const int lane = threadIdx.x & 31;
  const int mrow = lane & 15;
  const int kh   = lane >> 4;
  const int node = tileN * 16 + mrow;

  const float* Ap0 = A0 + (size_t)node * NFEAT + 2 * kh;
  v8f c[8];
#pragma unroll
  for (int t = 0; t < 8; ++t) c[t] = (v8f){};

  // K-half 1: columns [0,128) from A0
  for (int k0 = 0; k0 < NFEAT; k0 += 4) {
    const int ka = k0 + 2 * kh;
    const v2f a = *reinterpret_cast<const v2f*>(Ap0 + k0);
    const float* Wr = W + (size_t)ka * NFEAT + mrow;
#pragma unroll
    for (int t = 0; t < 8; ++t) {
      v2f b;
      b.x = Wr[t * 16];
      b.y = Wr[NFEAT + t * 16];
      c[t] = wmma_f32x4(a, b, c[t]);
    }
  }
  // K-half 2: columns [128,K) from A1 (skipped entirely when K == 128)
  if (A1 != nullptr) {
    const float* Ap1 = A1 + (size_t)node * NFEAT + 2 * kh;
    for (int k0 = NFEAT; k0 < K; k0 += 4) {
      const int ka = k0 + 2 * kh;
      const v2f a = *reinterpret_cast<const v2f*>(Ap1 + (k0 - NFEAT));
      const float* Wr = W + (size_t)ka * NFEAT + mrow;
#pragma unroll
      for (int t = 0; t < 8; ++t) {
        v2f b;
        b.x = Wr[t * 16];
        b.y = Wr[NFEAT + t * 16];
        c[t] = wmma_f32x4(a, b, c[t]);
      }
    }
  }

#pragma unroll
  for (int t = 0; t < 8; ++t) {
    const float bv = bias[t * 16 + mrow];
#pragma unroll
    for (int v = 0; v < 8; ++v) {
      const int m = tileN * 16 + v + 8 * kh;
      out[(size_t)m * NFEAT + t * 16 + mrow] = c[t][v] + bv;
    }
  }
}

__global__ void zero_kernel(float* __restrict__ p, long long n) {
  const long long i = (long long)blockIdx.x * blockDim.x + threadIdx.x;
  if (i < n) p[i] = 0.0f;
}

// --------------------------------------------------------------------------
// y[g_row[e]] += g_val[e] * x[g_col[e]]  (segment_sum).  One wave per edge,
// lane handles 4 feats (float4 gather, 4x global_atomic_add_f32 scatter).
// --------------------------------------------------------------------------
__global__ void spmm_scatter_kernel(const float* __restrict__ x,
                                    const float* __restrict__ gval,
                                    const int* __restrict__ grow,
                                    const int* __restrict__ gcol,
                                    float* __restrict__ y,
                                    int nedges) {
  const int e = global_wave();
  if (e >= nedges) return;
  const int lane = threadIdx.x & 31;
  const int r  = grow[e];
  const int ci = gcol[e];
  const float v = gval[e];
  const float4 xv = *reinterpret_cast<const float4*>(x + (size_t)ci * NFEAT + lane * 4);
  float* yp = y + (size_t)r * NFEAT + lane * 4;
  atomic_add_f32(yp + 0, v * xv.x);
  atomic_add_f32(yp + 1, v * xv.y);
  atomic_add_f32(yp + 2, v * xv.z);
  atomic_add_f32(yp + 3, v * xv.w);
}

// --------------------------------------------------------------------------
// out[i] = src[idx[i]] / max(||src[idx[i]]||, eps) — l2norm fused into gather.
// --------------------------------------------------------------------------
__global__ void gather_l2norm_kernel(const float* __restrict__ src,
                                     const int* __restrict__ idx,
                                     float* __restrict__ out,
                                     int n) {
  const int row = global_wave();
  if (row >= n) return;
  const int lane = threadIdx.x & 31;
  const int node = idx[row];
  const float4 xv = *reinterpret_cast<const float4*>(src + (size_t)node * NFEAT + lane * 4);
  float ss = xv.x * xv.x + xv.y * xv.y + xv.z * xv.z + xv.w * xv.w;
  for (int off = 16; off > 0; off >>= 1) ss += __shfl_xor(ss, off, 32);
  const float rinv = 1.0f / fmaxf(sqrtf(ss), EPSN);
  float4 o;
  o.x = xv.x * rinv; o.y = xv.y * rinv; o.z = xv.z * rinv; o.w = xv.w * rinv;
  *reinterpret_cast<float4*>(out + (size_t)row * NFEAT + lane * 4) = o;
}

__global__ void finalize_loss_kernel(const float* __restrict__ acc,
                                     float* __restrict__ out) {
  if (threadIdx.x == 0 && blockIdx.x == 0) {
    const float inv = 1.0f / ((float)NATTRI * (float)NNODES);
    out[0] = BETA_POS * acc[0] * inv + BETA_NEG * acc[1] * inv;
  }
}

// --------------------------------------------------------------------------
extern "C" void kernel_launch(void* const* d_in, const int* in_sizes, int n_in,
                              void* d_out, int out_size, void* d_ws, size_t ws_size,
                              hipStream_t stream) {
  const float* emb_attr = (const float*)d_in[0];
  const float* emb_node = (const float*)d_in[1];
  const float* W0       = (const float*)d_in[2];
  const float* b0       = (const float*)d_in[3];
  const float* W1       = (const float*)d_in[4];
  const float* b1       = (const float*)d_in[5];
  const float* H        = (const float*)d_in[6];
  const float* g_val    = (const float*)d_in[7];
  const int*   g_row    = (const int*)d_in[8];
  const int*   g_col    = (const int*)d_in[9];
  const int*   pos_src  = (const int*)d_in[10];
  const int*   pos_dst  = (const int*)d_in[11];
  const int*   neg_src  = (const int*)d_in[12];
  const int*   neg_dst  = (const int*)d_in[13];
  const int nedges = in_sizes[7];
  const int epos   = in_sizes[10];

  float* out = (float*)d_out;

  // Workspace: 2 loss accumulators (padded) + two ping-pong N x 128 buffers.
  // bufA: agg -> y0 -> y1 ; bufB: x0 -> x1.   (~102.4 MB total)
  float* acc  = (float*)d_ws;
  float* bufA = acc + 64;
  float* bufB = bufA + (size_t)NNODES * NFEAT;

  const int TB  = 256;             // 8 wave32 per block
  const int WPB = TB / 32;
  const long long nfloats = (long long)NNODES * NFEAT;
  const int zblocks = (int)((nfloats + TB - 1) / TB);
  const int rowWaves = NNODES / 16;                            // 6250
  const int rowBlocks = (rowWaves + WPB - 1) / WPB;            // 782

  // 0) clear loss accumulators (must be re-zeroed every call)
  zero_kernel<<<1, 64, 0, stream>>>(acc, 2);

  // 1) reconstruction loss (independent of the GNN path)
  recon_loss_kernel<<<rowBlocks, TB, 0, stream>>>(emb_attr, emb_node, H, acc);

  // 2) agg = H^T @ emb_attr -> bufA
  agg_kernel<<<rowBlocks, TB, 0, stream>>>(H, emb_attr, bufA);

  // 3) x0 = [emb_node | agg] @ W0 + b0 -> bufB   (concat fused via K split)
  linear_kernel<<<rowBlocks, TB, 0, stream>>>(emb_node, bufA, W0, b0, bufB,
                                              2 * NFEAT);

  // 4) y0 = spmm(x0) -> bufA (agg dead; zero then scatter)
  zero_kernel<<<zblocks, TB, 0, stream>>>(bufA, nfloats);
  spmm_scatter_kernel<<<(nedges + WPB - 1) / WPB, TB, 0, stream>>>(
      bufB, g_val, g_row, g_col, bufA, nedges);

  // 5) layer-0 output gathers: e0 = l2norm(y0) computed on the fly
  const size_t stride = (size_t)epos * NFEAT;    // one (layer, array) chunk
  const int gblocks = (epos + WPB - 1) / WPB;
  gather_l2norm_kernel<<<gblocks, TB, 0, stream>>>(bufA, pos_src, out + 0 * 2 * stride, epos);
  gather_l2norm_kernel<<<gblocks, TB, 0, stream>>>(bufA, pos_dst, out + 1 * 2 * stride, epos);
  gather_l2norm_kernel<<<gblocks, TB, 0, stream>>>(bufA, neg_src, out + 2 * 2 * stride, epos);
  gather_l2norm_kernel<<<gblocks, TB, 0, stream>>>(bufA, neg_dst, out + 3 * 2 * stride, epos);

  // 6) x1 = y0 @ W1 + b1 -> bufB (note: un-normalized y0, as in reference)
  linear_kernel<<<rowBlocks, TB, 0, stream>>>(bufA, nullptr, W1, b1, bufB, NFEAT);

  // 7) y1 = spmm(x1) -> bufA (y0 dead after steps 5-6; stream order protects)
  zero_kernel<<<zblocks, TB, 0, stream>>>(bufA, nfloats);
  spmm_scatter_kernel<<<(nedges + WPB - 1) / WPB, TB, 0, stream>>>(
      bufB, g_val, g_row, g_col, bufA, nedges);

  // 8) layer-1 output gathers: e1 = l2norm(y1)
  gather_l2norm_kernel<<<gblocks, TB, 0, stream>>>(bufA, pos_src, out + 0 * 2 * stride + stride, epos);
  gather_l2norm_kernel<<<gblocks, TB, 0, stream>>>(bufA, pos_dst, out + 1 * 2 * stride + stride, epos);
  gather_l2norm_kernel<<<gblocks, TB, 0, stream>>>(bufA, neg_src, out + 2 * 2 * stride + stride, epos);
  gather_l2norm_kernel<<<gblocks, TB, 0, stream>>>(bufA, neg_dst, out + 3 * 2 * stride + stride, epos);

  // 9) scalar loss at the very end of d_out
  finalize_loss_kernel<<<1, 32, 0, stream>>>(acc, out + 8 * stride);
}